// GINModel_15058155340592
// MI455X (gfx1250) — compile-verified
//
#include <hip/hip_runtime.h>
#include <hip/hip_bf16.h>

#define N_NODES   100000
#define N_EDGES   1600000
#define NUM_GRAPHS 1024
#define IN_FEAT   7
#define HDIM      128
#define CAT       (3 * HDIM)     // 384
#define CLF_H     (2 * HDIM)     // 256
#define NUM_CLASSES 2
#define BN_EPS    1e-5f

typedef float v2f __attribute__((ext_vector_type(2)));
typedef float v8f __attribute__((ext_vector_type(8)));

// ---------------------------------------------------------------- utilities
__global__ void zero_f32(float* __restrict__ p, long long n) {
    long long i = (long long)blockIdx.x * blockDim.x + threadIdx.x;
    if (i < n) p[i] = 0.0f;
}

// w[K][N] -> wt[N][K]
__global__ void transpose_f32(const float* __restrict__ w, float* __restrict__ wt,
                              int K, int N) {
    int i = blockIdx.x * blockDim.x + threadIdx.x;
    if (i >= K * N) return;
    int k = i / N;
    int n = i - k * N;
    wt[n * K + k] = w[i];
}

// generic scatter-add (used for layer 0, F = 7)
__global__ void scatter_add(const float* __restrict__ h, int ldh,
                            const int* __restrict__ src, const int* __restrict__ dst,
                            float* __restrict__ agg, int lda,
                            int F, long long total) {
    long long i = (long long)blockIdx.x * blockDim.x + threadIdx.x;
    if (i >= total) return;
    long long e = i / F;
    int f = (int)(i - e * (long long)F);
    float v = h[(long long)src[e] * ldh + f];
    atomicAdd(&agg[(long long)dst[e] * lda + f], v);
}

// vectorized scatter-add for F = 128: thread = (edge, 4 contiguous feats)
__global__ void scatter_add_128(const float* __restrict__ h, int ldh,
                                const int* __restrict__ src, const int* __restrict__ dst,
                                float* __restrict__ agg, long long total /* E*32 */) {
    long long i = (long long)blockIdx.x * blockDim.x + threadIdx.x;
    if (i >= total) return;
    int e = (int)(i >> 5);
    int f = (int)(i & 31) << 2;
    const float4 v = *(const float4*)(h + (long long)src[e] * ldh + f);
    float* base = agg + (long long)dst[e] * HDIM + f;
    atomicAdd(base + 0, v.x);
    atomicAdd(base + 1, v.y);
    atomicAdd(base + 2, v.z);
    atomicAdd(base + 3, v.w);
}

// layer 0 first MLP: z[n,j] = relu(b1[j] + sum_k (x[n,k]+agg[n,k]) * w1[k,j]), K=7
__global__ void layer0_mlp1(const float* __restrict__ x, const float* __restrict__ agg,
                            const float* __restrict__ w1, const float* __restrict__ b1,
                            float* __restrict__ z) {
    int i = blockIdx.x * blockDim.x + threadIdx.x;   // n*128 + j
    if (i >= N_NODES * HDIM) return;
    int n = i >> 7;
    int j = i & 127;
    float acc = b1[j];
#pragma unroll
    for (int k = 0; k < IN_FEAT; ++k)
        acc += (x[n * IN_FEAT + k] + agg[n * IN_FEAT + k]) * w1[k * HDIM + j];
    z[i] = fmaxf(acc, 0.0f);
}

// ---------------------------------------------------------------- WMMA GEMM
// C[M x N] = epilogue( (A (+ A2)) [M x K] @ W [K x N] + bias )
// W pre-transposed as WT[N][K] so B fragments are contiguous b64 loads.
// One wave computes a 16(M) x 64(N) tile: 4 chained V_WMMA_F32_16X16X4_F32 per
// K-step sharing one A fragment. Inner loop is software-pipelined: the loads
// for step k+4 are issued before the WMMAs of step k, so waits drain to a
// partial loadcnt instead of zero. M%16==0, N%64==0, K%8==0, blockDim=256.
template <bool HAS_A2, bool HAS_BN, bool RELU>
__global__ void wmma_gemm_f32(const float* __restrict__ A,  int lda,
                              const float* __restrict__ A2, int lda2,
                              const float* __restrict__ WT,
                              const float* __restrict__ bias,
                              const float* __restrict__ bnScale,
                              const float* __restrict__ bnShift,
                              float* __restrict__ C, int ldc,
                              int M, int N, int K) {
    const int tilesN     = N >> 6;
    const int tilesTotal = (M >> 4) * tilesN;
    const int waveId = blockIdx.x * (blockDim.x >> 5) + (threadIdx.x >> 5);
    if (waveId >= tilesTotal) return;                 // wave-uniform: EXEC stays all-1s
    const int tileM = waveId / tilesN;
    const int tileN = waveId - tileM * tilesN;

    const int lane    = threadIdx.x & 31;
    const int halfSel = lane >> 4;                    // 0 -> K{0,1}, 1 -> K{2,3}
    const int l15     = lane & 15;
    const int koff    = halfSel << 1;

    const float* aRow  = A + (long long)(tileM * 16 + l15) * lda;
    const float* a2Row = HAS_A2 ? A2 + (long long)(tileM * 16 + l15) * lda2 : nullptr;
    const int colBase = tileN * 64 + l15;
    const float* bCol0 = WT + (long long)(colBase +  0) * K;
    const float* bCol1 = WT + (long long)(colBase + 16) * K;
    const float* bCol2 = WT + (long long)(colBase + 32) * K;
    const float* bCol3 = WT + (long long)(colBase + 48) * K;

    v8f acc0 = {0.f,0.f,0.f,0.f,0.f,0.f,0.f,0.f};
    v8f acc1 = acc0, acc2 = acc0, acc3 = acc0;

    // --- pipeline prologue: fragments for k = 0 ---
    v2f a_p  = *(const v2f*)(aRow  + koff);
    v2f a2_p = {0.f, 0.f};
    if constexpr (HAS_A2) a2_p = *(const v2f*)(a2Row + koff);
    v2f b0_p = *(const v2f*)(bCol0 + koff);
    v2f b1_p = *(const v2f*)(bCol1 + koff);
    v2f b2_p = *(const v2f*)(bCol2 + koff);
    v2f b3_p = *(const v2f*)(bCol3 + koff);

#pragma unroll 2
    for (int k = 4; k < K; k += 4) {
        const int kn = k + koff;
        // stage fragments for the NEXT step (loads fly during this step's WMMAs)
        v2f a_n  = *(const v2f*)(aRow  + kn);
        v2f a2_n = {0.f, 0.f};
        if constexpr (HAS_A2) a2_n = *(const v2f*)(a2Row + kn);
        v2f b0_n = *(const v2f*)(bCol0 + kn);
        v2f b1_n = *(const v2f*)(bCol1 + kn);
        v2f b2_n = *(const v2f*)(bCol2 + kn);
        v2f b3_n = *(const v2f*)(bCol3 + kn);

        v2f a_use = a_p;
        if constexpr (HAS_A2) a_use += a2_p;
        acc0 = __builtin_amdgcn_wmma_f32_16x16x4_f32(false, a_use, false, b0_p, (short)0, acc0, false, false);
        acc1 = __builtin_amdgcn_wmma_f32_16x16x4_f32(false, a_use, false, b1_p, (short)0, acc1, false, false);
        acc2 = __builtin_amdgcn_wmma_f32_16x16x4_f32(false, a_use, false, b2_p, (short)0, acc2, false, false);
        acc3 = __builtin_amdgcn_wmma_f32_16x16x4_f32(false, a_use, false, b3_p, (short)0, acc3, false, false);

        a_p = a_n; a2_p = a2_n;
        b0_p = b0_n; b1_p = b1_n; b2_p = b2_n; b3_p = b3_n;
    }

    // --- pipeline epilogue: last step ---
    {
        v2f a_use = a_p;
        if constexpr (HAS_A2) a_use += a2_p;
        acc0 = __builtin_amdgcn_wmma_f32_16x16x4_f32(false, a_use, false, b0_p, (short)0, acc0, false, false);
        acc1 = __builtin_amdgcn_wmma_f32_16x16x4_f32(false, a_use, false, b1_p, (short)0, acc1, false, false);
        acc2 = __builtin_amdgcn_wmma_f32_16x16x4_f32(false, a_use, false, b2_p, (short)0, acc2, false, false);
        acc3 = __builtin_amdgcn_wmma_f32_16x16x4_f32(false, a_use, false, b3_p, (short)0, acc3, false, false);
    }

    v8f accs[4] = {acc0, acc1, acc2, acc3};
#pragma unroll
    for (int t = 0; t < 4; ++t) {
        const int col = tileN * 64 + t * 16 + l15;
        const float bc = bias ? bias[col] : 0.0f;
        float sc = 1.0f, sh = 0.0f;
        if constexpr (HAS_BN) { sc = bnScale[col]; sh = bnShift[col]; }
#pragma unroll
        for (int r = 0; r < 8; ++r) {
            const int row = tileM * 16 + r + (halfSel << 3);
            float v = accs[t][r] + bc;
            if constexpr (HAS_BN) v = v * sc + sh;
            if constexpr (RELU)   v = fmaxf(v, 0.0f);
            C[(long long)row * ldc + col] = v;
        }
    }
}

// ---------------------------------------------------------------- pooling / classifier tail
__global__ void pool_add4(const float* __restrict__ hcat, const int* __restrict__ batch,
                          float* __restrict__ g, long long total /* N*(CAT/4) */) {
    long long i = (long long)blockIdx.x * blockDim.x + threadIdx.x;
    if (i >= total) return;
    long long n = i / (CAT / 4);
    int f = (int)(i - n * (CAT / 4)) << 2;
    const float4 v = *(const float4*)(hcat + n * CAT + f);
    float* base = g + (long long)batch[n] * CAT + f;
    atomicAdd(base + 0, v.x);
    atomicAdd(base + 1, v.y);
    atomicAdd(base + 2, v.z);
    atomicAdd(base + 3, v.w);
}

__global__ void bn_prep(const float* __restrict__ gamma, const float* __restrict__ beta,
                        const float* __restrict__ mean,  const float* __restrict__ var,
                        float* __restrict__ s, float* __restrict__ t) {
    int j = blockIdx.x * blockDim.x + threadIdx.x;
    if (j >= CLF_H) return;
    float sc = gamma[j] * rsqrtf(var[j] + BN_EPS);
    s[j] = sc;
    t[j] = beta[j] - mean[j] * sc;
}

__global__ void clf_final(const float* __restrict__ z1, const float* __restrict__ w2,
                          const float* __restrict__ b2, float* __restrict__ out) {
    int i = blockIdx.x * blockDim.x + threadIdx.x;   // g*2 + c
    if (i >= NUM_GRAPHS * NUM_CLASSES) return;
    int gi = i >> 1;
    int c  = i & 1;
    float acc = b2[c];
    for (int k = 0; k < CLF_H; ++k)
        acc += z1[gi * CLF_H + k] * w2[k * NUM_CLASSES + c];
    out[i] = acc;
}

// ---------------------------------------------------------------- launch
static inline int cdiv(long long a, long long b) { return (int)((a + b - 1) / b); }

extern "C" void kernel_launch(void* const* d_in, const int* in_sizes, int n_in,
                              void* d_out, int out_size, void* d_ws, size_t ws_size,
                              hipStream_t stream) {
    const float* x     = (const float*)d_in[0];
    const int*   ei    = (const int*)d_in[1];
    const int*   src   = ei;
    const int*   dst   = ei + N_EDGES;
    const int*   batch = (const int*)d_in[2];

    const float* w1[3] = { (const float*)d_in[3],  (const float*)d_in[7],  (const float*)d_in[11] };
    const float* b1[3] = { (const float*)d_in[4],  (const float*)d_in[8],  (const float*)d_in[12] };
    const float* w2[3] = { (const float*)d_in[5],  (const float*)d_in[9],  (const float*)d_in[13] };
    const float* b2[3] = { (const float*)d_in[6],  (const float*)d_in[10], (const float*)d_in[14] };

    const float* clf_w1 = (const float*)d_in[15];
    const float* clf_b1 = (const float*)d_in[16];
    const float* clf_w2 = (const float*)d_in[17];
    const float* clf_b2 = (const float*)d_in[18];
    const float* bn_g   = (const float*)d_in[19];
    const float* bn_b   = (const float*)d_in[20];
    const float* bn_m   = (const float*)d_in[21];
    const float* bn_v   = (const float*)d_in[22];

    float* out = (float*)d_out;

    // workspace layout (floats)
    float* agg   = (float*)d_ws;                                   // N*128
    float* ztmp  = agg   + (long long)N_NODES * HDIM;              // N*128
    float* hcat  = ztmp  + (long long)N_NODES * HDIM;              // N*384
    float* gbuf  = hcat  + (long long)N_NODES * CAT;               // 1024*384
    float* z1    = gbuf  + (long long)NUM_GRAPHS * CAT;            // 1024*256
    float* bnS   = z1    + (long long)NUM_GRAPHS * CLF_H;          // 256
    float* bnT   = bnS   + CLF_H;                                  // 256
    float* wt2_0 = bnT   + CLF_H;                                  // 128*128
    float* wt1_1 = wt2_0 + HDIM * HDIM;
    float* wt2_1 = wt1_1 + HDIM * HDIM;
    float* wt1_2 = wt2_1 + HDIM * HDIM;
    float* wt2_2 = wt1_2 + HDIM * HDIM;
    float* wtc   = wt2_2 + HDIM * HDIM;                            // 256*384

    const int TB = 256;

    // ---------------- weight transposes (WT[N][K]) ----------------
    transpose_f32<<<cdiv(HDIM * HDIM, TB), TB, 0, stream>>>(w2[0], wt2_0, HDIM, HDIM);
    transpose_f32<<<cdiv(HDIM * HDIM, TB), TB, 0, stream>>>(w1[1], wt1_1, HDIM, HDIM);
    transpose_f32<<<cdiv(HDIM * HDIM, TB), TB, 0, stream>>>(w2[1], wt2_1, HDIM, HDIM);
    transpose_f32<<<cdiv(HDIM * HDIM, TB), TB, 0, stream>>>(w1[2], wt1_2, HDIM, HDIM);
    transpose_f32<<<cdiv(HDIM * HDIM, TB), TB, 0, stream>>>(w2[2], wt2_2, HDIM, HDIM);
    transpose_f32<<<cdiv(CAT * CLF_H, TB), TB, 0, stream>>>(clf_w1, wtc, CAT, CLF_H);

    const float* wt1[3] = { nullptr, wt1_1, wt1_2 };
    const float* wt2[3] = { wt2_0,   wt2_1, wt2_2 };

    const int nodeTiles = (N_NODES / 16) * (HDIM / 64);   // 16x64 tiles per wave

    // ---------------- layer 0 (K=7 scatter + scalar MLP1 + WMMA MLP2) ----------------
    {
        long long nAgg = (long long)N_NODES * IN_FEAT;
        zero_f32<<<cdiv(nAgg, TB), TB, 0, stream>>>(agg, nAgg);
        long long tot = (long long)N_EDGES * IN_FEAT;
        scatter_add<<<cdiv(tot, TB), TB, 0, stream>>>(x, IN_FEAT, src, dst, agg, IN_FEAT, IN_FEAT, tot);
        layer0_mlp1<<<cdiv((long long)N_NODES * HDIM, TB), TB, 0, stream>>>(x, agg, w1[0], b1[0], ztmp);
        // h0 = relu(ztmp @ w2_0 + b2_0) -> hcat[:, 0:128]
        wmma_gemm_f32<false, false, true><<<cdiv(nodeTiles, 8), TB, 0, stream>>>(
            ztmp, HDIM, nullptr, 0, wt2[0], b2[0], nullptr, nullptr,
            hcat + 0, CAT, N_NODES, HDIM, HDIM);
    }

    // ---------------- layers 1, 2 (full WMMA path) ----------------
    for (int l = 1; l < 3; ++l) {
        const float* hin = hcat + (l - 1) * HDIM;                  // row stride CAT
        long long nAgg = (long long)N_NODES * HDIM;
        zero_f32<<<cdiv(nAgg, TB), TB, 0, stream>>>(agg, nAgg);
        long long tot = (long long)N_EDGES * 32;                   // (edge, 4-feat) threads
        scatter_add_128<<<cdiv(tot, TB), TB, 0, stream>>>(hin, CAT, src, dst, agg, tot);

        // z = relu((h + agg) @ w1 + b1)
        wmma_gemm_f32<true, false, true><<<cdiv(nodeTiles, 8), TB, 0, stream>>>(
            hin, CAT, agg, HDIM, wt1[l], b1[l], nullptr, nullptr,
            ztmp, HDIM, N_NODES, HDIM, HDIM);
        // h = relu(z @ w2 + b2) -> hcat[:, l*128:(l+1)*128]
        wmma_gemm_f32<false, false, true><<<cdiv(nodeTiles, 8), TB, 0, stream>>>(
            ztmp, HDIM, nullptr, 0, wt2[l], b2[l], nullptr, nullptr,
            hcat + l * HDIM, CAT, N_NODES, HDIM, HDIM);
    }

    // ---------------- global add pool ----------------
    {
        long long nG = (long long)NUM_GRAPHS * CAT;
        zero_f32<<<cdiv(nG, TB), TB, 0, stream>>>(gbuf, nG);
        long long tot = (long long)N_NODES * (CAT / 4);
        pool_add4<<<cdiv(tot, TB), TB, 0, stream>>>(hcat, batch, gbuf, tot);
    }

    // ---------------- classifier ----------------
    bn_prep<<<1, CLF_H, 0, stream>>>(bn_g, bn_b, bn_m, bn_v, bnS, bnT);
    {
        // z1 = relu(BN(g @ clf_w1 + clf_b1)), M=1024, N=256, K=384
        int tiles = (NUM_GRAPHS / 16) * (CLF_H / 64);
        wmma_gemm_f32<false, true, true><<<cdiv(tiles, 8), TB, 0, stream>>>(
            gbuf, CAT, nullptr, 0, wtc, clf_b1, bnS, bnT,
            z1, CLF_H, NUM_GRAPHS, CLF_H, CAT);
    }
    clf_final<<<cdiv(NUM_GRAPHS * NUM_CLASSES, TB), TB, 0, stream>>>(z1, clf_w2, clf_b2, out);
}